// ARMA_20383914787287
// MI455X (gfx1250) — compile-verified
//
#include <hip/hip_runtime.h>
#include <hip/hip_bf16.h>

typedef __attribute__((ext_vector_type(2))) float v2f;
typedef __attribute__((ext_vector_type(8))) float v8f;

#define LDS_STRIDE 272  // 256 + 16 pad: rows kb and kb+2 land on disjoint bank halves

// ---------------------------------------------------------------------------
// 1) degree accumulation: deg[col[e]] += 1
// ---------------------------------------------------------------------------
__global__ __launch_bounds__(256) void deg_kernel(const long long* __restrict__ col,
                                                  float* __restrict__ deg, long long E) {
    long long e = (long long)blockIdx.x * blockDim.x + threadIdx.x;
    if (e < E) atomicAdd(&deg[col[e]], 1.0f);
}

// 2) dinv[n] = deg>0 ? deg^-0.5 : 0   (in place)
__global__ __launch_bounds__(256) void dinv_kernel(float* __restrict__ deg, int n) {
    int i = blockIdx.x * blockDim.x + threadIdx.x;
    if (i < n) {
        float d = deg[i];
        deg[i] = (d > 0.0f) ? rsqrtf(d) : 0.0f;
    }
}

// 3) norm[e] = dinv[row[e]] * dinv[col[e]]
__global__ __launch_bounds__(256) void norm_kernel(const long long* __restrict__ row,
                                                   const long long* __restrict__ col,
                                                   const float* __restrict__ dinv,
                                                   float* __restrict__ norm, long long E) {
    long long e = (long long)blockIdx.x * blockDim.x + threadIdx.x;
    if (e < E) norm[e] = dinv[row[e]] * dinv[col[e]];
}

// ---------------------------------------------------------------------------
// 4) pack weights: Wpack[f][c], c = [init k0 | init k1 | root k0 | root k1]
//    init_w/root_w are [K, Fin, 64] row-major
// ---------------------------------------------------------------------------
__global__ __launch_bounds__(256) void pack_weights(const float* __restrict__ initw,
                                                    const float* __restrict__ rootw,
                                                    float* __restrict__ wpack, int fin) {
    int idx = blockIdx.x * blockDim.x + threadIdx.x;
    int total = fin * 256;
    if (idx >= total) return;
    int f = idx >> 8;
    int c = idx & 255;
    float v;
    if (c < 128) {
        int k = c >> 6, o = c & 63;
        v = initw[(size_t)k * fin * 64 + (size_t)f * 64 + o];
    } else {
        int c2 = c - 128;
        int k = c2 >> 6, o = c2 & 63;
        v = rootw[(size_t)k * fin * 64 + (size_t)f * 64 + o];
    }
    wpack[(size_t)f * 256 + c] = v;
}

// ---------------------------------------------------------------------------
// 5) WMMA GEMM: Cout[N x 256] = A[N x fin] @ Wpack[fin x 256]
//    One block = 16 rows. 8 waves; each wave owns two 16-wide column tiles.
//    f32 WMMA 16x16x4. B staged in LDS (fin * LDS_STRIDE floats, dynamic,
//    row-padded so the two lane-halves of each ds load hit disjoint banks).
//    Fragment layouts per CDNA5 ISA 7.12.2:
//      A  (16x4):  lane l, vgpr j  -> A[l%16][2*(l/16)+j]
//      B  (4x16):  lane l, vgpr j  -> B[2*(l/16)+j][l%16]
//      C/D(16x16): lane l, vgpr i  -> C[8*(l/16)+i][l%16]
// ---------------------------------------------------------------------------
__global__ __launch_bounds__(256) void gemm_wmma_f32(const float* __restrict__ A,
                                                     const float* __restrict__ Wpack,
                                                     float* __restrict__ Cout,
                                                     int n_rows, int fin) {
    extern __shared__ float ldsB[]; // fin * LDS_STRIDE floats

    const int tid = threadIdx.x;
    // stage B into LDS with float4 loads (padded row stride)
    {
        const int total4 = (fin * 256) >> 2;
        for (int i = tid; i < total4; i += 256) {
            int f = (i >> 6);         // i*4 / 256
            int c = (i & 63) << 2;    // (i*4) % 256
            const float4 v = *(const float4*)(Wpack + (size_t)f * 256 + c);
            *(float4*)(ldsB + f * LDS_STRIDE + c) = v;
        }
    }
    __syncthreads();

    const int wave = tid >> 5;
    const int lane = tid & 31;
    const int lmod = lane & 15;
    const int lhi  = lane >> 4;          // 0 or 1

    const int m0 = blockIdx.x * 16;
    int row = m0 + lmod;
    if (row >= n_rows) row = n_rows - 1; // select-based clamp, no EXEC divergence

    const float* __restrict__ arow = A + (size_t)row * fin + 2 * lhi;
    const int n0 = wave * 32;            // two column tiles: n0, n0+16

    v8f acc0 = {};
    v8f acc1 = {};

#pragma unroll 4
    for (int kk = 0; kk < fin; kk += 4) {
        v2f a;
        a.x = arow[kk];
        a.y = arow[kk + 1];

        const int kb = kk + 2 * lhi;
        v2f b0, b1;
        b0.x = ldsB[(kb    ) * LDS_STRIDE + n0 + lmod];
        b0.y = ldsB[(kb + 1) * LDS_STRIDE + n0 + lmod];
        b1.x = ldsB[(kb    ) * LDS_STRIDE + n0 + 16 + lmod];
        b1.y = ldsB[(kb + 1) * LDS_STRIDE + n0 + 16 + lmod];

        acc0 = __builtin_amdgcn_wmma_f32_16x16x4_f32(
            false, a, false, b0, (short)0, acc0, false, false);
        acc1 = __builtin_amdgcn_wmma_f32_16x16x4_f32(
            false, a, false, b1, (short)0, acc1, false, false);
    }

#pragma unroll
    for (int i = 0; i < 8; ++i) {
        int r = m0 + 8 * lhi + i;
        if (r < n_rows) {
            Cout[(size_t)r * 256 + n0 + lmod]      = acc0[i];
            Cout[(size_t)r * 256 + n0 + 16 + lmod] = acc1[i];
        }
    }
}

// ---------------------------------------------------------------------------
// 6) edge scatter: one wave per edge; lanes cover 128 cols (K=2 x 64) as float4
//    agg[col*128 + c] += HR[row*256 + c] * norm[e],  c in [0,128)
// ---------------------------------------------------------------------------
__global__ __launch_bounds__(256) void edge_scatter(const long long* __restrict__ rowi,
                                                    const long long* __restrict__ coli,
                                                    const float* __restrict__ norm,
                                                    const float* __restrict__ HR,
                                                    float* __restrict__ agg, long long E) {
    long long gid = (long long)blockIdx.x * blockDim.x + threadIdx.x;
    long long e = gid >> 5;
    int lane = (int)(gid & 31);
    if (e >= E) return;

    long long r = rowi[e];
    long long c = coli[e];
    float nv = norm[e];

    const float4 v = *(const float4*)(HR + (size_t)r * 256 + lane * 4);
    float* dst = agg + (size_t)c * 128 + lane * 4;
    atomicAdd(dst + 0, v.x * nv);
    atomicAdd(dst + 1, v.y * nv);
    atomicAdd(dst + 2, v.z * nv);
    atomicAdd(dst + 3, v.w * nv);
}

// ---------------------------------------------------------------------------
// 7) combine: out[n,o] = mean_k relu(agg[k] + root[k] + bias[k])
// ---------------------------------------------------------------------------
__global__ __launch_bounds__(256) void combine_kernel(const float* __restrict__ agg,
                                                      const float* __restrict__ HR,
                                                      const float* __restrict__ bias,
                                                      float* __restrict__ out, int n_nodes) {
    int idx = blockIdx.x * blockDim.x + threadIdx.x;
    if (idx >= n_nodes * 64) return;
    int n = idx >> 6;
    int o = idx & 63;
    float v0 = agg[(size_t)n * 128 + o]      + HR[(size_t)n * 256 + 128 + o] + bias[o];
    float v1 = agg[(size_t)n * 128 + 64 + o] + HR[(size_t)n * 256 + 192 + o] + bias[64 + o];
    v0 = fmaxf(v0, 0.0f);
    v1 = fmaxf(v1, 0.0f);
    out[idx] = 0.5f * (v0 + v1);
}

// ---------------------------------------------------------------------------
extern "C" void kernel_launch(void* const* d_in, const int* in_sizes, int n_in,
                              void* d_out, int out_size, void* d_ws, size_t ws_size,
                              hipStream_t stream) {
    const float*     x        = (const float*)d_in[0];
    const long long* edge     = (const long long*)d_in[1];
    const float*     init_w1  = (const float*)d_in[2];
    const float*     root_w1  = (const float*)d_in[3];
    const float*     b1       = (const float*)d_in[4];
    const float*     init_w2  = (const float*)d_in[5];
    const float*     root_w2  = (const float*)d_in[6];
    const float*     b2       = (const float*)d_in[7];
    float*           out      = (float*)d_out;

    const int       N   = in_sizes[0] / 128;
    const long long E   = in_sizes[1] / 2;
    const long long* row = edge;
    const long long* col = edge + E;

    // workspace layout (256B-aligned)
    auto align256 = [](size_t v) { return (v + 255) & ~(size_t)255; };
    char* ws = (char*)d_ws;
    size_t off = 0;
    float* dinv  = (float*)(ws + off); off = align256(off + (size_t)N * 4);
    float* norm  = (float*)(ws + off); off = align256(off + (size_t)E * 4);
    float* wpack = (float*)(ws + off); off = align256(off + (size_t)128 * 256 * 4);
    float* HR    = (float*)(ws + off); off = align256(off + (size_t)N * 256 * 4);
    float* agg   = (float*)(ws + off); off = align256(off + (size_t)N * 128 * 4);
    float* h1    = (float*)(ws + off); off = align256(off + (size_t)N * 64 * 4);
    (void)ws_size;

    // idempotent, host-side, capture-safe; no static guard (harness requirement)
    hipFuncSetAttribute((const void*)gemm_wmma_f32,
                        hipFuncAttributeMaxDynamicSharedMemorySize,
                        128 * LDS_STRIDE * 4);

    const unsigned eBlocks  = (unsigned)((E + 255) / 256);
    const unsigned weBlocks = (unsigned)((E * 32 + 255) / 256);
    const unsigned mTiles   = (unsigned)((N + 15) / 16);
    const unsigned noBlocks = (unsigned)(((size_t)N * 64 + 255) / 256);

    // ---- norm ----
    hipMemsetAsync(dinv, 0, (size_t)N * 4, stream);
    deg_kernel<<<eBlocks, 256, 0, stream>>>(col, dinv, E);
    dinv_kernel<<<(N + 255) / 256, 256, 0, stream>>>(dinv, N);
    norm_kernel<<<eBlocks, 256, 0, stream>>>(row, col, dinv, norm, E);

    // ---- layer 1 ----
    pack_weights<<<(128 * 256 + 255) / 256, 256, 0, stream>>>(init_w1, root_w1, wpack, 128);
    gemm_wmma_f32<<<mTiles, 256, 128 * LDS_STRIDE * 4, stream>>>(x, wpack, HR, N, 128);
    hipMemsetAsync(agg, 0, (size_t)N * 128 * 4, stream);
    edge_scatter<<<weBlocks, 256, 0, stream>>>(row, col, norm, HR, agg, E);
    combine_kernel<<<noBlocks, 256, 0, stream>>>(agg, HR, b1, h1, N);

    // ---- layer 2 ----
    pack_weights<<<(64 * 256 + 255) / 256, 256, 0, stream>>>(init_w2, root_w2, wpack, 64);
    gemm_wmma_f32<<<mTiles, 256, 64 * LDS_STRIDE * 4, stream>>>(h1, wpack, HR, N, 64);
    hipMemsetAsync(agg, 0, (size_t)N * 128 * 4, stream);
    edge_scatter<<<weBlocks, 256, 0, stream>>>(row, col, norm, HR, agg, E);
    combine_kernel<<<noBlocks, 256, 0, stream>>>(agg, HR, b2, out, N);
}